// ObsNegativeSamplingLoss_30743375904836
// MI455X (gfx1250) — compile-verified
//
#include <hip/hip_runtime.h>
#include <hip/hip_bf16.h>
#include <stdint.h>

typedef __attribute__((ext_vector_type(2))) float        v2f;
typedef __attribute__((ext_vector_type(8))) float        v8f;
typedef __attribute__((ext_vector_type(4))) unsigned int u32x4;
typedef __attribute__((ext_vector_type(8))) int          i32x8;
typedef __attribute__((ext_vector_type(4))) int          i32x4;

#define TOTAL  100000
#define BATCH  512
#define NBINS  7
#define NSMALL 4
#define NBIG   3
#define NSAMP  20

// merge-scale for online softmax; returns exp(m-mn) but is safe when both are -inf
__device__ __forceinline__ float merge_scale(float m, float mn) {
  return (m == mn) ? 1.0f : __expf(m - mn);
}

__device__ __forceinline__ unsigned hash32(unsigned x) {
  x ^= x >> 16; x *= 0x7feb352dU;
  x ^= x >> 15; x *= 0x846ca68bU;
  x ^= x >> 16;
  return x;
}

// ---------------------------------------------------------------------------
// Kernel A: small bins (nb <= 1000). One 128-thread block per (row, bin).
// TDM (tensor_load_to_lds) stages the 1 x nb fp32 slice into LDS once; both
// the online-softmax pass and the target lookup then hit LDS.
// ---------------------------------------------------------------------------
__global__ void small_bin_kernel(const float* __restrict__ logits,
                                 const unsigned char* __restrict__ mask,
                                 const int* __restrict__ targets,
                                 float* __restrict__ partials) {
  __shared__ float smem[1024];
  __shared__ float redM[4], redS[4];
  const int kNb[NSMALL]    = {100, 200, 500, 1000};
  const int kStart[NSMALL] = {0, 100, 300, 800};

  const int row   = blockIdx.x;
  const int bin   = blockIdx.y;
  const int nb    = kNb[bin];
  const int start = kStart[bin];
  const float* src = logits + (size_t)row * TOTAL + start;

  if (threadIdx.x == 0) {
    // Build Tensor DMA Descriptor (D#): 2D tile nb x 1, data_size = 4 bytes.
    unsigned long long ga = (unsigned long long)(const void*)src;  // 57-bit global addr
    unsigned lds = (unsigned)(unsigned long long)(void*)smem;      // LDS byte offset

    u32x4 g0;
    g0.x = 1u;                                                // count=1, user mode
    g0.y = lds;                                               // lds_addr
    g0.z = (unsigned)ga;                                      // global_addr[31:0]
    g0.w = ((unsigned)(ga >> 32) & 0x01FFFFFFu) | (2u << 30); // addr[56:32] | type=2

    i32x8 g1;
    g1[0] = (int)(2u << 16);                              // data_size=2 (4B), no mcast
    g1[1] = (int)(((unsigned)nb & 0xFFFFu) << 16);        // tensor_dim0 lo16
    g1[2] = (int)((((unsigned)nb >> 16) & 0xFFFFu) |      // tensor_dim0 hi16
                  (1u << 16));                            // tensor_dim1 lo16 = 1
    g1[3] = (int)(((unsigned)nb & 0xFFFFu) << 16);        // tensor_dim1 hi=0 | tile_dim0=nb
    g1[4] = 1;                                            // tile_dim1=1, tile_dim2=0
    g1[5] = nb;                                           // tensor_dim0_stride lo32
    g1[6] = 0;                                            // stride0 hi16 | stride1 lo16
    g1[7] = 0;                                            // stride1 hi32

    i32x4 z4 = {0, 0, 0, 0};
#if __clang_major__ >= 23
    i32x8 z8 = {0, 0, 0, 0, 0, 0, 0, 0};
    __builtin_amdgcn_tensor_load_to_lds(g0, g1, z4, z4, z8, 0);
#else
    __builtin_amdgcn_tensor_load_to_lds(g0, g1, z4, z4, 0);
#endif
    __builtin_amdgcn_s_wait_tensorcnt((short)0);
  }
  __syncthreads();

  // Online softmax (max m, scaled sum s) over the LDS-resident slice.
  float m = -INFINITY, s = 0.0f;
  for (int i = threadIdx.x; i < nb; i += blockDim.x) {
    float x  = smem[i];
    float mn = fmaxf(m, x);
    s = s * merge_scale(m, mn) + __expf(x - mn);
    m = mn;
  }
  // intra-wave (wave32) reduction
  for (int off = 16; off >= 1; off >>= 1) {
    float mo = __shfl_xor(m, off);
    float so = __shfl_xor(s, off);
    float mn = fmaxf(m, mo);
    s = s * merge_scale(m, mn) + so * merge_scale(mo, mn);
    m = mn;
  }
  const int wave = threadIdx.x >> 5;
  const int lane = threadIdx.x & 31;
  if (lane == 0) { redM[wave] = m; redS[wave] = s; }
  __syncthreads();

  if (threadIdx.x == 0) {
    float M = redM[0], S = redS[0];
    for (int w = 1; w < 4; ++w) {
      float mo = redM[w], so = redS[w];
      float mn = fmaxf(M, mo);
      S = S * merge_scale(M, mn) + so * merge_scale(mo, mn);
      M = mn;
    }
    float logZ = M + __logf(S);
    int   t    = targets[row * NBINS + bin];
    float xt   = smem[t];
    unsigned char msk = mask[(size_t)row * TOTAL + start + t];  // obs = !mask
    partials[bin * BATCH + row] = msk ? 0.0f : (logZ - xt);
  }
}

// ---------------------------------------------------------------------------
// Kernel B: big bins (sampled softmax). One wave per (row, bin).
// Lane 0 = target logit; lanes 1..20 rejection-sample a uniform observed
// index (counter-hash RNG, deterministic); then a 21-way wave softmax.
// NOTE: RNG stream differs from jax.random.categorical (threefry) by design.
// ---------------------------------------------------------------------------
__global__ void big_bin_kernel(const float* __restrict__ logits,
                               const unsigned char* __restrict__ mask,
                               const int* __restrict__ targets,
                               float* __restrict__ partials) {
  const int kNb[NBIG]    = {5000, 20000, 73200};
  const int kStart[NBIG] = {1800, 6800, 26800};

  const int row   = blockIdx.x;
  const int b     = blockIdx.y;
  const int nb    = kNb[b];
  const int start = kStart[b];
  const float*         lrow = logits + (size_t)row * TOTAL + start;
  const unsigned char* mrow = mask   + (size_t)row * TOTAL + start;
  const int lane = threadIdx.x;

  float val;
  if (lane == 0) {
    int t = targets[row * NBINS + NSMALL + b];
    val = lrow[t];
  } else if (lane <= NSAMP) {
    // rejection sampling: uniform over observed (mask==0) entries, ~2 tries
    unsigned seed = hash32(0x9E3779B9u * (unsigned)(row * NBIG + b + 1)) ^
                    ((unsigned)lane * 0x85EBCA6Bu);
    unsigned idx = 0;
    for (int a = 0; a < 64; ++a) {
      idx = hash32(seed ^ ((unsigned)a * 0xC2B2AE35u)) % (unsigned)nb;
      if (!mrow[idx]) break;
    }
    val = lrow[idx];
  } else {
    val = -INFINITY;  // idle lanes contribute nothing
  }

  float m = val;
  for (int off = 16; off >= 1; off >>= 1) m = fmaxf(m, __shfl_xor(m, off));
  float e = (lane <= NSAMP) ? __expf(val - m) : 0.0f;
  float ssum = e;
  for (int off = 16; off >= 1; off >>= 1) ssum += __shfl_xor(ssum, off);

  float target_val = __shfl(val, 0);
  if (lane == 0)
    partials[NSMALL * BATCH + b * BATCH + row] = (m + __logf(ssum)) - target_val;
}

// ---------------------------------------------------------------------------
// Kernel C: deterministic final sum of 3584 partials with a WMMA-assisted
// cross-lane fold. A = ones(16x4), B VGPR0 = per-lane accumulators -> each
// column of D holds acc[n] + acc[n+16]; 4 shfl_xor steps finish the sum.
// ---------------------------------------------------------------------------
__global__ void final_reduce_kernel(const float* __restrict__ partials,
                                    float* __restrict__ out) {
  const int lane = threadIdx.x;
  const int N = (NSMALL + NBIG) * BATCH;
  float acc = 0.0f;
  for (int i = lane; i < N; i += 32) acc += partials[i];  // fixed order

  v2f a;  a.x = 1.0f; a.y = 1.0f;   // A = ones(16x4)
  v2f bb; bb.x = acc; bb.y = 0.0f;  // B VGPR0 = lane accumulators
  v8f c = {};
  v8f d = __builtin_amdgcn_wmma_f32_16x16x4_f32(
      /*neg_a=*/false, a, /*neg_b=*/false, bb,
      /*c_mod=*/(short)0, c, /*reuse_a=*/false, /*reuse_b=*/false);

  float v = d[0];                                  // acc[n%16] + acc[n%16+16]
  for (int off = 8; off >= 1; off >>= 1) v += __shfl_xor(v, off);

  if (lane == 0) {
    out[0] = v;                                           // loss
    out[1] = v * (1.0f / (512.0f * 0.69314718055994531f)); // loss/(B*ln2)
  }
}

extern "C" void kernel_launch(void* const* d_in, const int* in_sizes, int n_in,
                              void* d_out, int out_size, void* d_ws, size_t ws_size,
                              hipStream_t stream) {
  (void)in_sizes; (void)n_in; (void)out_size; (void)ws_size;
  const int*           targets = (const int*)d_in[0];
  const float*         logits  = (const float*)d_in[1];
  const unsigned char* mask    = (const unsigned char*)d_in[2];  // jnp bool = 1 byte
  float* partials = (float*)d_ws;   // 3584 floats, each written exactly once
  float* out      = (float*)d_out;

  small_bin_kernel<<<dim3(BATCH, NSMALL), 128, 0, stream>>>(logits, mask, targets, partials);
  big_bin_kernel  <<<dim3(BATCH, NBIG),    32, 0, stream>>>(logits, mask, targets, partials);
  final_reduce_kernel<<<1, 32, 0, stream>>>(partials, out);
}